// GraphSAGEBackbone_74878459838735
// MI455X (gfx1250) — compile-verified
//
#include <hip/hip_runtime.h>

#define FIN  128
#define FHID 128
#define FOUT 64

typedef __attribute__((ext_vector_type(2))) float v2f;
typedef __attribute__((ext_vector_type(8))) float v8f;

// ---------------- utility kernels ----------------

__global__ void zero_kernel(float* __restrict__ p, int n) {
  int i = blockIdx.x * blockDim.x + threadIdx.x;
  if (i < n) p[i] = 0.0f;
}

__global__ void edge_count_kernel(const int* __restrict__ ei, float* __restrict__ cnt, int E) {
  int e = blockIdx.x * blockDim.x + threadIdx.x;
  if (e < E) atomicAdd(&cnt[ei[E + e]], 1.0f);
}

__global__ void inv_count_kernel(float* __restrict__ cnt, int n) {
  int i = blockIdx.x * blockDim.x + threadIdx.x;
  if (i < n) {
    float c = cnt[i];
    cnt[i] = 1.0f / (c > 1.0f ? c : 1.0f);
  }
}

// one wave32 per edge; lane handles 4 consecutive features (128 total)
__global__ void edge_scatter_kernel(const int* __restrict__ ei,
                                    const float* __restrict__ feat,
                                    float* __restrict__ acc, int E) {
  int wid = blockIdx.x * (blockDim.x >> 5) + (threadIdx.x >> 5);
  if (wid >= E) return;
  int lane = threadIdx.x & 31;
  int src = ei[wid];
  int dst = ei[E + wid];
  float4 v = ((const float4*)(feat + (size_t)src * FIN))[lane];
  float* d = acc + (size_t)dst * FIN + lane * 4;
  atomicAdd(d + 0, v.x);
  atomicAdd(d + 1, v.y);
  atomicAdd(d + 2, v.z);
  atomicAdd(d + 3, v.w);
}

// 64-wide scatter: each half-wave handles one edge, lane covers 4 floats
__global__ void edge_scatter64_kernel(const int* __restrict__ ei,
                                      const float* __restrict__ q,
                                      float* __restrict__ acc, int E) {
  int wid = blockIdx.x * (blockDim.x >> 5) + (threadIdx.x >> 5);
  int lane = threadIdx.x & 31;
  int e = wid * 2 + (lane >> 4);
  if (e >= E) return;
  int l = lane & 15;
  int src = ei[e];
  int dst = ei[E + e];
  float4 v = ((const float4*)(q + (size_t)src * FOUT))[l];
  float* d = acc + (size_t)dst * FOUT + l * 4;
  atomicAdd(d + 0, v.x);
  atomicAdd(d + 1, v.y);
  atomicAdd(d + 2, v.z);
  atomicAdd(d + 3, v.w);
}

// ---------------- layer 1: h1 = (sum*inv_cnt) @ Wl + x @ Wr + b ----------------
// block = 256 threads = 8 waves; block covers 16 node rows x 128 output cols
__global__ __launch_bounds__(256) void sage_gemm1_kernel(
    const float* __restrict__ x, const float* __restrict__ sum,
    const float* __restrict__ inv_cnt,
    const float* __restrict__ Wl, const float* __restrict__ Wr,
    const float* __restrict__ bias, float* __restrict__ h1) {
  __shared__ float aggT[16 * 132];
  __shared__ float xT[16 * 132];
  const int t = threadIdx.x;
  const int m0 = blockIdx.x * 16;

  for (int f = t; f < 16 * 32; f += 256) {
    int row = f >> 5;
    int c4 = f & 31;
    int node = m0 + row;
    float4 xs = ((const float4*)(x + (size_t)node * FIN))[c4];
    float4 ss = ((const float4*)(sum + (size_t)node * FIN))[c4];
    float ic = inv_cnt[node];
    float* xr = &xT[row * 132 + c4 * 4];
    xr[0] = xs.x; xr[1] = xs.y; xr[2] = xs.z; xr[3] = xs.w;
    float* ar = &aggT[row * 132 + c4 * 4];
    ar[0] = ss.x * ic; ar[1] = ss.y * ic; ar[2] = ss.z * ic; ar[3] = ss.w * ic;
  }
  __syncthreads();

  const int wave = t >> 5, lane = t & 31;
  const int half = lane >> 4, l = lane & 15;
  const int n0 = wave * 16;

  float bv = bias[n0 + l];
  v8f c;
#pragma unroll
  for (int v = 0; v < 8; ++v) c[v] = bv;

#pragma unroll 8
  for (int k0 = 0; k0 < FIN; k0 += 4) {
    int kr = k0 + 2 * half;
    v2f aA = *(const v2f*)&aggT[l * 132 + kr];
    v2f aX = *(const v2f*)&xT[l * 132 + kr];
    v2f bL, bR;
    bL[0] = Wl[(kr + 0) * FHID + n0 + l];
    bL[1] = Wl[(kr + 1) * FHID + n0 + l];
    bR[0] = Wr[(kr + 0) * FHID + n0 + l];
    bR[1] = Wr[(kr + 1) * FHID + n0 + l];
    c = __builtin_amdgcn_wmma_f32_16x16x4_f32(false, aA, false, bL, (short)0, c, false, false);
    c = __builtin_amdgcn_wmma_f32_16x16x4_f32(false, aX, false, bR, (short)0, c, false, false);
  }
#pragma unroll
  for (int v = 0; v < 8; ++v)
    h1[(size_t)(m0 + v + 8 * half) * FHID + n0 + l] = c[v];
}

// ---------------- batchnorm (training stats) ----------------

__global__ void bn_stats_kernel(const float* __restrict__ h1, float* __restrict__ stats, int N) {
  int t = threadIdx.x;
  int c = t & 127;
  int r0 = blockIdx.x * 256 + (t >> 7);
  int rend = blockIdx.x * 256 + 256;
  if (rend > N) rend = N;
  float s = 0.f, s2 = 0.f;
  for (int r = r0; r < rend; r += 2) {
    float v = h1[(size_t)r * FHID + c];
    s += v;
    s2 += v * v;
  }
  atomicAdd(&stats[c], s);
  atomicAdd(&stats[128 + c], s2);
}

__global__ void bn_finalize_kernel(float* __restrict__ stats, int N) {
  int c = threadIdx.x;  // 0..127
  float inv_n = 1.0f / (float)N;
  float mean = stats[c] * inv_n;
  float var = stats[128 + c] * inv_n - mean * mean;
  stats[256 + c] = mean;
  stats[384 + c] = rsqrtf(var + 1e-5f);
}

__global__ void bn_relu_kernel(float* __restrict__ h1, const float* __restrict__ stats,
                               const float* __restrict__ gamma, const float* __restrict__ beta,
                               int total) {
  int i = blockIdx.x * blockDim.x + threadIdx.x;
  if (i < total) {
    int c = i & 127;
    float v = (h1[i] - stats[256 + c]) * stats[384 + c] * gamma[c] + beta[c];
    h1[i] = v > 0.f ? v : 0.f;
  }
}

// ---------------- layer 2 (optimized): project h BEFORE aggregation ----------------
// q = h @ Wl2 ; r = h @ Wr2 + b2     (agg is linear, so agg(h)@Wl2 == agg(h@Wl2))
// block = 256 threads = 8 waves; block covers 32 node rows x 64 output cols
__global__ __launch_bounds__(256) void gemm2_pre_kernel(
    const float* __restrict__ h,
    const float* __restrict__ Wl, const float* __restrict__ Wr,
    const float* __restrict__ bias,
    float* __restrict__ q, float* __restrict__ r) {
  __shared__ float hT[32 * 132];
  const int t = threadIdx.x;
  const int m0 = blockIdx.x * 32;

  for (int f = t; f < 32 * 32; f += 256) {
    int row = f >> 5;
    int c4 = f & 31;
    float4 xs = ((const float4*)(h + (size_t)(m0 + row) * FHID))[c4];
    float* xr = &hT[row * 132 + c4 * 4];
    xr[0] = xs.x; xr[1] = xs.y; xr[2] = xs.z; xr[3] = xs.w;
  }
  __syncthreads();

  const int wave = t >> 5, lane = t & 31;
  const int rbase = (wave >> 2) * 16;
  const int n0 = (wave & 3) * 16;
  const int half = lane >> 4, l = lane & 15;

  float bv = bias[n0 + l];
  v8f cq, cr;
#pragma unroll
  for (int v = 0; v < 8; ++v) { cq[v] = 0.f; cr[v] = bv; }

#pragma unroll 8
  for (int k0 = 0; k0 < FHID; k0 += 4) {
    int kr = k0 + 2 * half;
    v2f a = *(const v2f*)&hT[(rbase + l) * 132 + kr];   // shared A-frag for both GEMMs
    v2f bL, bR;
    bL[0] = Wl[(kr + 0) * FOUT + n0 + l];
    bL[1] = Wl[(kr + 1) * FOUT + n0 + l];
    bR[0] = Wr[(kr + 0) * FOUT + n0 + l];
    bR[1] = Wr[(kr + 1) * FOUT + n0 + l];
    cq = __builtin_amdgcn_wmma_f32_16x16x4_f32(false, a, false, bL, (short)0, cq, false, false);
    cr = __builtin_amdgcn_wmma_f32_16x16x4_f32(false, a, false, bR, (short)0, cr, false, false);
  }
#pragma unroll
  for (int v = 0; v < 8; ++v) {
    size_t off = (size_t)(m0 + rbase + v + 8 * half) * FOUT + n0 + l;
    q[off] = cq[v];
    r[off] = cr[v];
  }
}

__global__ void final_combine_kernel(const float* __restrict__ sum64,
                                     const float* __restrict__ r,
                                     const float* __restrict__ inv_cnt,
                                     float* __restrict__ out, int total) {
  int i = blockIdx.x * blockDim.x + threadIdx.x;
  if (i < total) {
    int n = i >> 6;
    out[i] = sum64[i] * inv_cnt[n] + r[i];
  }
}

// ---------------- layer 2 (fallback, 128-wide scatter then fused GEMM) ----------------
__global__ __launch_bounds__(256) void sage_gemm2_kernel(
    const float* __restrict__ h, const float* __restrict__ sum,
    const float* __restrict__ inv_cnt,
    const float* __restrict__ Wl, const float* __restrict__ Wr,
    const float* __restrict__ bias, float* __restrict__ out) {
  __shared__ float aggT[32 * 132];
  __shared__ float hT[32 * 132];
  const int t = threadIdx.x;
  const int m0 = blockIdx.x * 32;

  for (int f = t; f < 32 * 32; f += 256) {
    int row = f >> 5;
    int c4 = f & 31;
    int node = m0 + row;
    float4 xs = ((const float4*)(h + (size_t)node * FHID))[c4];
    float4 ss = ((const float4*)(sum + (size_t)node * FHID))[c4];
    float ic = inv_cnt[node];
    float* xr = &hT[row * 132 + c4 * 4];
    xr[0] = xs.x; xr[1] = xs.y; xr[2] = xs.z; xr[3] = xs.w;
    float* ar = &aggT[row * 132 + c4 * 4];
    ar[0] = ss.x * ic; ar[1] = ss.y * ic; ar[2] = ss.z * ic; ar[3] = ss.w * ic;
  }
  __syncthreads();

  const int wave = t >> 5, lane = t & 31;
  const int rbase = (wave >> 2) * 16;
  const int n0 = (wave & 3) * 16;
  const int half = lane >> 4, l = lane & 15;

  float bv = bias[n0 + l];
  v8f c;
#pragma unroll
  for (int v = 0; v < 8; ++v) c[v] = bv;

#pragma unroll 8
  for (int k0 = 0; k0 < FHID; k0 += 4) {
    int kr = k0 + 2 * half;
    v2f aA = *(const v2f*)&aggT[(rbase + l) * 132 + kr];
    v2f aX = *(const v2f*)&hT[(rbase + l) * 132 + kr];
    v2f bL, bR;
    bL[0] = Wl[(kr + 0) * FOUT + n0 + l];
    bL[1] = Wl[(kr + 1) * FOUT + n0 + l];
    bR[0] = Wr[(kr + 0) * FOUT + n0 + l];
    bR[1] = Wr[(kr + 1) * FOUT + n0 + l];
    c = __builtin_amdgcn_wmma_f32_16x16x4_f32(false, aA, false, bL, (short)0, c, false, false);
    c = __builtin_amdgcn_wmma_f32_16x16x4_f32(false, aX, false, bR, (short)0, c, false, false);
  }
#pragma unroll
  for (int v = 0; v < 8; ++v)
    out[(size_t)(m0 + rbase + v + 8 * half) * FOUT + n0 + l] = c[v];
}

// ---------------- launch ----------------

extern "C" void kernel_launch(void* const* d_in, const int* in_sizes, int n_in,
                              void* d_out, int out_size, void* d_ws, size_t ws_size,
                              hipStream_t stream) {
  (void)n_in; (void)out_size;
  const float* x      = (const float*)d_in[0];
  const int*   ei     = (const int*)d_in[1];
  const float* Wl1    = (const float*)d_in[2];
  const float* b1     = (const float*)d_in[3];
  const float* Wr1    = (const float*)d_in[4];
  const float* gamma1 = (const float*)d_in[5];
  const float* beta1  = (const float*)d_in[6];
  const float* Wl2    = (const float*)d_in[7];
  const float* b2     = (const float*)d_in[8];
  const float* Wr2    = (const float*)d_in[9];
  float* out = (float*)d_out;

  const int N = in_sizes[0] / FIN;  // 100000 (divisible by 32)
  const int E = in_sizes[1] / 2;    // 1600000
  const int NF = N * FHID;
  auto cdiv = [](int a, int b) { return (a + b - 1) / b; };

  // Optimized layout: regionA = [q (N*64) | r (N*64) | sum64 (N*64)];
  // layer-1 sumbuf (N*128) aliases [q|r] (dead before q/r are written).
  const size_t need_opt = ((size_t)N * 192 + (size_t)N * FHID + (size_t)N + 512) * sizeof(float);
  const bool opt = (ws_size >= need_opt);

  float* ws     = (float*)d_ws;
  float* sumbuf = ws;                                              // N*128
  float* h1     = ws + (opt ? (size_t)N * 192 : (size_t)N * FHID); // N*128
  float* cnt    = h1 + (size_t)N * FHID;                           // N
  float* stats  = cnt + N;                                         // 512

  // ---- layer 1 ----
  zero_kernel<<<cdiv(NF, 256), 256, 0, stream>>>(sumbuf, NF);
  zero_kernel<<<cdiv(N, 256), 256, 0, stream>>>(cnt, N);
  zero_kernel<<<2, 256, 0, stream>>>(stats, 512);

  edge_count_kernel<<<cdiv(E, 256), 256, 0, stream>>>(ei, cnt, E);
  inv_count_kernel<<<cdiv(N, 256), 256, 0, stream>>>(cnt, N);
  edge_scatter_kernel<<<cdiv(E, 8), 256, 0, stream>>>(ei, x, sumbuf, E);

  sage_gemm1_kernel<<<N / 16, 256, 0, stream>>>(x, sumbuf, cnt, Wl1, Wr1, b1, h1);

  bn_stats_kernel<<<cdiv(N, 256), 256, 0, stream>>>(h1, stats, N);
  bn_finalize_kernel<<<1, 128, 0, stream>>>(stats, N);
  bn_relu_kernel<<<cdiv(NF, 256), 256, 0, stream>>>(h1, stats, gamma1, beta1, NF);

  // ---- layer 2 ----
  if (opt) {
    float* q     = ws;                       // aliases sumbuf[0 .. N*64)
    float* r     = ws + (size_t)N * FOUT;    // aliases sumbuf[N*64 .. N*128)
    float* sum64 = ws + (size_t)N * 2 * FOUT;
    const int NO = N * FOUT;

    gemm2_pre_kernel<<<N / 32, 256, 0, stream>>>(h1, Wl2, Wr2, b2, q, r);
    zero_kernel<<<cdiv(NO, 256), 256, 0, stream>>>(sum64, NO);
    edge_scatter64_kernel<<<cdiv(E, 16), 256, 0, stream>>>(ei, q, sum64, E);
    final_combine_kernel<<<cdiv(NO, 256), 256, 0, stream>>>(sum64, r, cnt, out, NO);
  } else {
    zero_kernel<<<cdiv(NF, 256), 256, 0, stream>>>(sumbuf, NF);
    edge_scatter_kernel<<<cdiv(E, 8), 256, 0, stream>>>(ei, h1, sumbuf, E);
    sage_gemm2_kernel<<<N / 32, 256, 0, stream>>>(h1, sumbuf, cnt, Wl2, Wr2, b2, out);
  }
}